// RecognitionModel_89378269430451
// MI455X (gfx1250) — compile-verified
//
#include <hip/hip_runtime.h>
#include <math.h>

typedef __attribute__((ext_vector_type(2))) float v2f;
typedef __attribute__((ext_vector_type(8))) float v8f;

#define BDIM 32
#define TSTEPS 2000
#define NCHAIN 512  // N_MC * N_TRIALS

// ---------------------------------------------------------------------------
// WMMA f32 16x16x4 helpers (CDNA5 / gfx1250, wave32)
// ---------------------------------------------------------------------------
__device__ __forceinline__ v8f wmma_f32(v2f a, v2f b, v8f c) {
  // 8-arg form: (neg_a, A, neg_b, B, c_mod, C, reuse_a, reuse_b)
  return __builtin_amdgcn_wmma_f32_16x16x4_f32(false, a, false, b, (short)0, c,
                                               false, false);
}

// A operand: 16x4 f32 slice of row-major M (stride 32).
// lanes 0-15: M=0..15, VGPR0=K0, VGPR1=K1; lanes 16-31: VGPR0=K2, VGPR1=K3
__device__ __forceinline__ v2f frag_a(const float* M, int r0, int k0, int lane) {
  int m = lane & 15, kh = lane >> 4;
  v2f a;
  a.x = M[(r0 + m) * BDIM + k0 + 2 * kh];
  a.y = M[(r0 + m) * BDIM + k0 + 2 * kh + 1];
  return a;
}
// B operand: 4x16 f32 slice of row-major M (stride 32).
__device__ __forceinline__ v2f frag_b(const float* M, int k0, int c0, int lane) {
  int n = lane & 15, kh = lane >> 4;
  v2f b;
  b.x = M[(k0 + 2 * kh) * BDIM + c0 + n];
  b.y = M[(k0 + 2 * kh + 1) * BDIM + c0 + n];
  return b;
}
// B operand of M^T, reading row-major M.
__device__ __forceinline__ v2f frag_bT(const float* M, int k0, int c0, int lane) {
  int n = lane & 15, kh = lane >> 4;
  v2f b;
  b.x = M[(c0 + n) * BDIM + k0 + 2 * kh];
  b.y = M[(c0 + n) * BDIM + k0 + 2 * kh + 1];
  return b;
}
// C/D 16x16 f32: VGPR v -> row (v + 8*(lane>=16)), col lane&15
__device__ __forceinline__ void store_c(float* D, int r0, int c0, int lane, v8f c) {
  int n = lane & 15, mh = (lane >> 4) * 8;
#pragma unroll
  for (int v = 0; v < 8; ++v) D[(r0 + mh + v) * BDIM + c0 + n] = c[v];
}
__device__ __forceinline__ v8f load_c(const float* D, int r0, int c0, int lane) {
  int n = lane & 15, mh = (lane >> 4) * 8;
  v8f c;
#pragma unroll
  for (int v = 0; v < 8; ++v) c[v] = D[(r0 + mh + v) * BDIM + c0 + n];
  return c;
}

// 32x32 = 32x32 * 32x32 GEMM with 4 waves (wave w does 16x16 tile (w>>1, w&1)).
// YT=true uses Y transposed.
template <bool YT>
__device__ __forceinline__ void gemm32(float* dst, const float* X, const float* Y,
                                       int wave, int lane) {
  int r0 = (wave >> 1) * 16, c0 = (wave & 1) * 16;
  v8f acc = {};
#pragma unroll
  for (int ks = 0; ks < 8; ++ks) {
    v2f a = frag_a(X, r0, ks * 4, lane);
    v2f b = YT ? frag_bT(Y, ks * 4, c0, lane) : frag_b(Y, ks * 4, c0, lane);
    acc = wmma_f32(a, b, acc);
  }
  store_c(dst, r0, c0, lane, acc);
}

// ---------------------------------------------------------------------------
// In-place 32x32 Cholesky (lower; strict upper zeroed). Wave 0 only,
// lane i owns row i. LDS ops from one wave are in-order -> no barrier needed.
// ---------------------------------------------------------------------------
__device__ __forceinline__ void chol32(float* S, int tid) {
  if (tid < 32) {
    int lane = tid;
    for (int k = 0; k < 32; ++k) {
      float d = sqrtf(fmaxf(S[k * 32 + k], 1e-30f));
      float inv = 1.0f / d;
      float l = (lane > k) ? S[lane * 32 + k] * inv : 0.0f;
      if (lane == k) S[k * 32 + k] = d;
      if (lane > k) S[lane * 32 + k] = l;
      for (int j = k + 1; j <= lane; ++j) S[lane * 32 + j] -= l * S[j * 32 + k];
    }
    for (int j = lane + 1; j < 32; ++j) S[lane * 32 + j] = 0.0f;
  }
}

// P := P * (L L^T)^{-1}  (right cho-solve; lane r owns row r of P).
__device__ __forceinline__ void rchosolve32(float* P, const float* L, int tid) {
  if (tid < 32) {
    float* p = P + tid * 32;
    for (int j = 0; j < 32; ++j) {  // y = p * L^{-T}  (forward)
      float s = p[j];
      for (int i = 0; i < j; ++i) s -= p[i] * L[j * 32 + i];
      p[j] = s / L[j * 32 + j];
    }
    for (int j = 31; j >= 0; --j) {  // x = y * L^{-1}  (backward)
      float s = p[j];
      for (int i = j + 1; i < 32; ++i) s -= p[i] * L[i * 32 + j];
      p[j] = s / L[j * 32 + j];
    }
  }
}

// ---------------------------------------------------------------------------
// Phase A: covariance filter (serial scan over t). One workgroup, 4 waves.
// ---------------------------------------------------------------------------
__global__ void __launch_bounds__(128) k_cov_filter(
    const float* __restrict__ Ag, const float* __restrict__ Wg,
    const float* __restrict__ Qg, const float* __restrict__ Rg,
    const float* __restrict__ Sig0, float* Sf_out, float* Ks_out, float* Sd_ws,
    float* Sdc_ws) {
  __shared__ float cA[1024], cW[1024], cQ[1024], cR[1024];
  __shared__ float mSf[1024], mSd[1024], mSdc[1024], mT1[1024], mT2[1024],
      mM[1024], mP[1024];
  int tid = threadIdx.x, wave = tid >> 5, lane = tid & 31;
#pragma unroll
  for (int i = 0; i < 8; ++i) {
    int idx = tid + i * 128;
    cA[idx] = Ag[idx];
    cW[idx] = Wg[idx];
    cQ[idx] = Qg[idx];
    cR[idx] = Rg[idx];
    mSd[idx] = Sig0[idx];  // t=0 uses Sigma0 in place of Sd
  }
  __syncthreads();

  for (int t = 0; t < TSTEPS; ++t) {
    if (t > 0) {
      gemm32<false>(mT1, cA, mSf, wave, lane);  // T1 = A Sf
      __syncthreads();
      gemm32<true>(mT2, mT1, cA, wave, lane);  // T2 = A Sf A^T
      __syncthreads();
#pragma unroll
      for (int i = 0; i < 8; ++i) {  // Sd = sym(T2) + Q ; SdJ = Sd + 1e-6 I
        int idx = tid + i * 128;
        int r = idx >> 5, c = idx & 31;
        float v = 0.5f * (mT2[idx] + mT2[c * 32 + r]) + cQ[idx];
        mSd[idx] = v;
        Sd_ws[(long)(t - 1) * 1024 + idx] = v;
        mSdc[idx] = v + ((r == c) ? 1e-6f : 0.0f);
      }
      __syncthreads();
      chol32(mSdc, tid);
      __syncthreads();
#pragma unroll
      for (int i = 0; i < 8; ++i) {
        int idx = tid + i * 128;
        Sdc_ws[(long)(t - 1) * 1024 + idx] = mSdc[idx];
      }
    }
    __syncthreads();
    gemm32<false>(mT2, cW, mSd, wave, lane);  // T2 = W Sd
    __syncthreads();
    gemm32<true>(mM, mT2, cW, wave, lane);  // M = W Sd W^T
    __syncthreads();
    float jit = (t > 0) ? 1e-6f : 0.0f;  // t=0: no jitter (matches reference)
#pragma unroll
    for (int i = 0; i < 8; ++i) {
      int idx = tid + i * 128;
      int r = idx >> 5, c = idx & 31;
      mM[idx] += cR[idx] + ((r == c) ? jit : 0.0f);
    }
    __syncthreads();
    chol32(mM, tid);  // S = chol(R + W Sd W^T [+ jit])
    __syncthreads();
    gemm32<true>(mP, mSd, cW, wave, lane);  // P = Sd W^T
    __syncthreads();
    rchosolve32(mP, mM, tid);  // K = P (S S^T)^{-1}
    __syncthreads();
#pragma unroll
    for (int i = 0; i < 8; ++i) {
      int idx = tid + i * 128;
      Ks_out[(long)t * 1024 + idx] = mP[idx];
    }
    gemm32<false>(mT1, mP, mT2, wave, lane);  // T1 = K (W Sd)
    __syncthreads();
#pragma unroll
    for (int i = 0; i < 8; ++i) {  // Sf = sym(Sd - T1)
      int idx = tid + i * 128;
      int r = idx >> 5, c = idx & 31;
      float v = 0.5f * ((mSd[idx] - mT1[idx]) + (mSd[c * 32 + r] - mT1[c * 32 + r]));
      mSf[idx] = v;
      Sf_out[(long)t * 1024 + idx] = v;
    }
    __syncthreads();
  }
}

// ---------------------------------------------------------------------------
// Phase B: covariance smoother gains + Sig_tilde_chol. One block per t.
// ---------------------------------------------------------------------------
__global__ void __launch_bounds__(128) k_cov_smoother(
    const float* __restrict__ Ag, const float* __restrict__ Sf_in,
    const float* __restrict__ Sd_ws, const float* __restrict__ Sdc_ws,
    float* SigChol_out, float* Cs_out) {
  __shared__ float cA[1024], mSf[1024], mSd[1024], mL[1024], mP[1024],
      mT1[1024], mT2[1024];
  int t = blockIdx.x;
  int tid = threadIdx.x, wave = tid >> 5, lane = tid & 31;
#pragma unroll
  for (int i = 0; i < 8; ++i) {
    int idx = tid + i * 128;
    cA[idx] = Ag[idx];
    mSf[idx] = Sf_in[(long)t * 1024 + idx];
  }
  if (t < TSTEPS - 1) {
#pragma unroll
    for (int i = 0; i < 8; ++i) {
      int idx = tid + i * 128;
      mSd[idx] = Sd_ws[(long)t * 1024 + idx];
      mL[idx] = Sdc_ws[(long)t * 1024 + idx];
    }
    __syncthreads();
    gemm32<true>(mP, mSf, cA, wave, lane);  // P = Sf A^T
    __syncthreads();
    rchosolve32(mP, mL, tid);  // C = Sf A^T Sd^{-1}
    __syncthreads();
#pragma unroll
    for (int i = 0; i < 8; ++i) {
      int idx = tid + i * 128;
      Cs_out[(long)t * 1024 + idx] = mP[idx];
    }
    gemm32<false>(mT1, mP, mSd, wave, lane);  // C Sd
    __syncthreads();
    gemm32<true>(mT2, mT1, mP, wave, lane);  // C Sd C^T
    __syncthreads();
#pragma unroll
    for (int i = 0; i < 8; ++i) {  // St = sym(Sf - C Sd C^T) + 1e-4 I
      int idx = tid + i * 128;
      int r = idx >> 5, c = idx & 31;
      float v = 0.5f * ((mSf[idx] - mT2[idx]) + (mSf[c * 32 + r] - mT2[c * 32 + r]));
      mL[idx] = v + ((r == c) ? 1e-4f : 0.0f);
    }
  } else {  // last t: chol(Sf + 1e-4 I)
    __syncthreads();
#pragma unroll
    for (int i = 0; i < 8; ++i) {
      int idx = tid + i * 128;
      int r = idx >> 5, c = idx & 31;
      mL[idx] = mSf[idx] + ((r == c) ? 1e-4f : 0.0f);
    }
  }
  __syncthreads();
  chol32(mL, tid);
  __syncthreads();
#pragma unroll
  for (int i = 0; i < 8; ++i) {
    int idx = tid + i * 128;
    SigChol_out[(long)t * 1024 + idx] = mL[idx];
  }
}

// ---------------------------------------------------------------------------
// Phase C: mean filter. 512 independent chains; one wave owns 16 chains
// as the M-dim of 16x32 WMMA tiles. Serial over t, 3 GEMMs per step.
// ---------------------------------------------------------------------------
__global__ void __launch_bounds__(32) k_mean_filter(
    const float* __restrict__ Ag, const float* __restrict__ Wg,
    const float* __restrict__ mu0, const float* __restrict__ xhat,
    const float* __restrict__ Ks, float* mus_filt) {
  __shared__ float ldsMu[512], ldsMd[512], ldsR[512];
  int lane = threadIdx.x;
  int c0 = blockIdx.x * 16;  // first chain in this group
  v2f bAT[2][8], bWT[2][8];  // loop-invariant B fragments of A^T and W^T
#pragma unroll
  for (int tc = 0; tc < 2; ++tc)
#pragma unroll
    for (int ks = 0; ks < 8; ++ks) {
      bAT[tc][ks] = frag_bT(Ag, ks * 4, tc * 16, lane);
      bWT[tc][ks] = frag_bT(Wg, ks * 4, tc * 16, lane);
    }
  int n = lane & 15, mh = (lane >> 4) * 8;

  for (int t = 0; t < TSTEPS; ++t) {
    if (t == 0) {  // mu_d = mu0 (broadcast to all chains)
      float m0 = mu0[lane];
      for (int r = 0; r < 16; ++r) ldsMd[r * 32 + lane] = m0;
    } else {  // Mu_d = Mu A^T
#pragma unroll
      for (int tc = 0; tc < 2; ++tc) {
        v8f acc = {};
#pragma unroll
        for (int ks = 0; ks < 8; ++ks)
          acc = wmma_f32(frag_a(ldsMu, 0, ks * 4, lane), bAT[tc][ks], acc);
        store_c(ldsMd, 0, tc * 16, lane, acc);
      }
    }
    if (t + 1 < TSTEPS)
      __builtin_prefetch(Ks + (long)(t + 1) * 1024 + lane * 32, 0, 0);
    // resid = x_t - Mu_d W^T  (x_hat strided: (chain, d, t), t innermost)
#pragma unroll
    for (int tc = 0; tc < 2; ++tc) {
      v8f acc = {};
#pragma unroll
      for (int ks = 0; ks < 8; ++ks)
        acc = wmma_f32(frag_a(ldsMd, 0, ks * 4, lane), bWT[tc][ks], acc);
      int d = tc * 16 + n;
#pragma unroll
      for (int v = 0; v < 8; ++v) {
        long chain = c0 + mh + v;
        float xv = xhat[(chain * 32 + d) * (long)TSTEPS + t];
        ldsR[(mh + v) * 32 + d] = xv - acc[v];
      }
    }
    // Mu = Mu_d + resid K_t^T
    const float* Kt = Ks + (long)t * 1024;
#pragma unroll
    for (int tc = 0; tc < 2; ++tc) {
      v8f acc = load_c(ldsMd, 0, tc * 16, lane);
#pragma unroll
      for (int ks = 0; ks < 8; ++ks)
        acc = wmma_f32(frag_a(ldsR, 0, ks * 4, lane),
                       frag_bT(Kt, ks * 4, tc * 16, lane), acc);
      store_c(ldsMu, 0, tc * 16, lane, acc);
#pragma unroll
      for (int v = 0; v < 8; ++v)
        mus_filt[((long)t * NCHAIN + c0 + mh + v) * 32 + tc * 16 + n] = acc[v];
    }
  }
}

// ---------------------------------------------------------------------------
// Phase D: mean smoother (backward). mu_d recomputed as mus_filt[t] A^T.
// ---------------------------------------------------------------------------
__global__ void __launch_bounds__(32) k_mean_smoother(
    const float* __restrict__ Ag, const float* __restrict__ Cs,
    const float* __restrict__ mus_filt, float* mus_smooth) {
  __shared__ float ldsMn[512], ldsMf[512], ldsR[512];
  int lane = threadIdx.x;
  int c0 = blockIdx.x * 16;
  v2f bAT[2][8];
#pragma unroll
  for (int tc = 0; tc < 2; ++tc)
#pragma unroll
    for (int ks = 0; ks < 8; ++ks) bAT[tc][ks] = frag_bT(Ag, ks * 4, tc * 16, lane);
  int n = lane & 15, mh = (lane >> 4) * 8;

  // init: Mn = mus_filt[T-1]; mus_smooth[T-1] = Mn
  for (int r = 0; r < 16; ++r) {
    long gi = ((long)(TSTEPS - 1) * NCHAIN + c0 + r) * 32 + lane;
    float v = mus_filt[gi];
    ldsMn[r * 32 + lane] = v;
    mus_smooth[gi] = v;
  }

  for (int t = TSTEPS - 2; t >= 0; --t) {
    for (int r = 0; r < 16; ++r)
      ldsMf[r * 32 + lane] = mus_filt[((long)t * NCHAIN + c0 + r) * 32 + lane];
    const float* Ct = Cs + (long)t * 1024;
    // R = Mn - Mf A^T
#pragma unroll
    for (int tc = 0; tc < 2; ++tc) {
      v8f acc = {};
#pragma unroll
      for (int ks = 0; ks < 8; ++ks)
        acc = wmma_f32(frag_a(ldsMf, 0, ks * 4, lane), bAT[tc][ks], acc);
#pragma unroll
      for (int v = 0; v < 8; ++v) {
        int idx = (mh + v) * 32 + tc * 16 + n;
        ldsR[idx] = ldsMn[idx] - acc[v];
      }
    }
    // Mn' = Mf + R C_t^T
#pragma unroll
    for (int tc = 0; tc < 2; ++tc) {
      v8f acc = load_c(ldsMf, 0, tc * 16, lane);
#pragma unroll
      for (int ks = 0; ks < 8; ++ks)
        acc = wmma_f32(frag_a(ldsR, 0, ks * 4, lane),
                       frag_bT(Ct, ks * 4, tc * 16, lane), acc);
      store_c(ldsMn, 0, tc * 16, lane, acc);
#pragma unroll
      for (int v = 0; v < 8; ++v)
        mus_smooth[((long)t * NCHAIN + c0 + mh + v) * 32 + tc * 16 + n] = acc[v];
    }
    if (t > 0) __builtin_prefetch(Cs + (long)(t - 1) * 1024 + lane * 32, 0, 0);
  }
}

// ---------------------------------------------------------------------------
extern "C" void kernel_launch(void* const* d_in, const int* in_sizes, int n_in,
                              void* d_out, int out_size, void* d_ws,
                              size_t ws_size, hipStream_t stream) {
  (void)in_sizes; (void)n_in; (void)out_size; (void)ws_size;
  const float* A = (const float*)d_in[0];
  const float* W = (const float*)d_in[1];
  const float* Q = (const float*)d_in[2];
  const float* R = (const float*)d_in[3];
  const float* Sig0 = (const float*)d_in[4];
  const float* mu0 = (const float*)d_in[5];
  const float* xhat = (const float*)d_in[6];

  float* out = (float*)d_out;
  const long SZ_MU = (long)TSTEPS * NCHAIN * 32;  // mus_* size
  const long SZ_MAT = (long)TSTEPS * 1024;
  float* mus_smooth = out;
  float* mus_filt = out + SZ_MU;
  float* sig_chol = out + 2 * SZ_MU;
  float* Sf_out = sig_chol + SZ_MAT;
  float* Ks_out = Sf_out + SZ_MAT;
  float* Cs_out = Ks_out + SZ_MAT;  // (T-1)*1024

  float* Sd_ws = (float*)d_ws;                       // (T-1)*1024
  float* Sdc_ws = Sd_ws + (long)(TSTEPS - 1) * 1024; // (T-1)*1024

  k_cov_filter<<<1, 128, 0, stream>>>(A, W, Q, R, Sig0, Sf_out, Ks_out, Sd_ws,
                                      Sdc_ws);
  k_cov_smoother<<<TSTEPS, 128, 0, stream>>>(A, Sf_out, Sd_ws, Sdc_ws, sig_chol,
                                             Cs_out);
  k_mean_filter<<<NCHAIN / 16, 32, 0, stream>>>(A, W, mu0, xhat, Ks_out,
                                                mus_filt);
  k_mean_smoother<<<NCHAIN / 16, 32, 0, stream>>>(A, Cs_out, mus_filt,
                                                  mus_smooth);
}